// _SlidingWindowCost_33638183862962
// MI455X (gfx1250) — compile-verified
//
#include <hip/hip_runtime.h>
#include <math.h>

// SlidingWindowCost for MI455X (gfx1250, wave32).
// Purely elementwise quaternion residuals -> HBM-bandwidth-bound (~300MB total
// traffic, ~13us floor at 23.3 TB/s). fp32 VALU path; LDS-staged coalesced
// output stores; b128 loads for quaternions; global_prefetch_b8 for streaming.

namespace {

constexpr float kEps = 1e-8f;
#define THREADS 256

struct V3 { float x, y, z; };
struct Q  { float x, y, z, w; };

__device__ __forceinline__ V3 mk3(float x, float y, float z) { return {x, y, z}; }
__device__ __forceinline__ V3 operator+(V3 a, V3 b) { return {a.x + b.x, a.y + b.y, a.z + b.z}; }
__device__ __forceinline__ V3 operator-(V3 a, V3 b) { return {a.x - b.x, a.y - b.y, a.z - b.z}; }
__device__ __forceinline__ V3 operator*(float s, V3 a) { return {s * a.x, s * a.y, s * a.z}; }

__device__ __forceinline__ float dot3(V3 a, V3 b) {
  return fmaf(a.x, b.x, fmaf(a.y, b.y, a.z * b.z));
}
__device__ __forceinline__ V3 cross3(V3 a, V3 b) {
  return { fmaf(a.y, b.z, -a.z * b.y),
           fmaf(a.z, b.x, -a.x * b.z),
           fmaf(a.x, b.y, -a.y * b.x) };
}

__device__ __forceinline__ Q qinv(Q q) { return {-q.x, -q.y, -q.z, q.w}; }

// qmul per reference: v = w1*v2 + w2*v1 + cross(v1,v2); w = w1*w2 - dot(v1,v2)
__device__ __forceinline__ Q qmul(Q a, Q b) {
  V3 v1{a.x, a.y, a.z}, v2{b.x, b.y, b.z};
  V3 c = cross3(v1, v2);
  V3 v = a.w * v2 + b.w * v1 + c;
  float w = fmaf(a.w, b.w, -dot3(v1, v2));
  return {v.x, v.y, v.z, w};
}

// qact(q, t) = t + 2*cross(v, cross(v,t) + w*t)
__device__ __forceinline__ V3 qact(Q q, V3 t) {
  V3 v{q.x, q.y, q.z};
  V3 inner = cross3(v, t) + q.w * t;
  return t + 2.0f * cross3(v, inner);
}

// qlog per reference (sign-fixed, theta = 2*atan2(|v|, w))
__device__ __forceinline__ V3 qlog(Q q) {
  float sgn = (q.w >= 0.0f) ? 1.0f : -1.0f;
  V3 v = sgn * mk3(q.x, q.y, q.z);
  float w = sgn * q.w;
  float n = sqrtf(dot3(v, v));
  float theta = 2.0f * atan2f(n, w);
  float coef = (n < kEps) ? 2.0f : theta / fmaxf(n, kEps);
  return coef * v;
}

__device__ __forceinline__ V3 load3(const float* __restrict__ p, int i) {
  const float* q = p + 3ll * i;
  return {q[0], q[1], q[2]};
}
__device__ __forceinline__ Q load4(const float* __restrict__ p, int i) {
  // quaternion arrays are 16B-element, base 256B-aligned -> global_load_b128
  const float4 f = reinterpret_cast<const float4*>(p)[i];
  return {f.x, f.y, f.z, f.w};
}

} // namespace

__global__ __launch_bounds__(THREADS)
void sliding_window_cost_kernel(
    const float* __restrict__ R0,        const float* __restrict__ t0,
    const float* __restrict__ rotations, const float* __restrict__ translations,
    const float* __restrict__ velocities,const float* __restrict__ bias_a,
    const float* __restrict__ bias_g,    const float* __restrict__ dR,
    const float* __restrict__ dv,        const float* __restrict__ dp,
    const float* __restrict__ dt,        const float* __restrict__ vR,
    const float* __restrict__ vt,        float* __restrict__ out, int N)
{
  // Stride-27 (odd) per-thread slots: conflict-free across 64 LDS banks.
  __shared__ float lds[THREADS * 27];

  const int tid  = threadIdx.x;
  const int base = blockIdx.x * THREADS;
  const int i    = base + tid;

  if (i < N) {
    // Prefetch streaming quaternion arrays one block ahead (global_prefetch_b8).
    const int pf = i + THREADS;
    if (pf < N) {
      __builtin_prefetch(rotations + 4ll * pf, 0, 1);
      __builtin_prefetch(dR        + 4ll * pf, 0, 1);
      __builtin_prefetch(vR        + 4ll * pf, 0, 1);
    }

    const Q  R_k    = (i == 0) ? load4(R0, 0) : load4(rotations, i - 1);
    const V3 t_k    = (i == 0) ? load3(t0, 0) : load3(translations, i - 1);
    const Q  R_kp1  = load4(rotations, i);
    const V3 t_kp1  = load3(translations, i);
    const V3 v_k    = load3(velocities, i);
    const V3 v_kp1  = load3(velocities, i + 1);
    const V3 ba_k   = load3(bias_a, i);
    const V3 ba_kp1 = load3(bias_a, i + 1);
    const V3 bg_k   = load3(bias_g, i);
    const V3 bg_kp1 = load3(bias_g, i + 1);
    const Q  dRi    = load4(dR, i);
    const V3 dvi    = load3(dv, i);
    const V3 dpi    = load3(dp, i);
    const float dti = dt[i];
    const Q  vRi    = load4(vR, i);
    const V3 vti    = load3(vt, i);

    const Q  Rk_inv    = qinv(R_k);
    const Q  rel_R_est = qmul(Rk_inv, R_kp1);
    const V3 dtrans    = t_kp1 - t_k;
    const V3 rel_t_est = qact(Rk_inv, dtrans);

    const V3 r_rot   = qlog(qmul(qinv(dRi), rel_R_est)) + dti * bg_k;
    const V3 r_cross = qact(Rk_inv, dtrans - dti * v_k) - dpi + (0.5f * dti * dti) * ba_k;
    const V3 r_dv    = v_kp1 - v_k - qact(R_k, dvi) + dti * ba_k;
    const V3 r_vrot  = qlog(qmul(qinv(vRi), rel_R_est));
    const V3 r_vt    = rel_t_est - vti;
    const V3 d_ba    = ba_kp1 - ba_k;
    const V3 d_bg    = bg_kp1 - bg_k;

    // All lambdas are 1.0 -> residuals written as-is, reference row order.
    float* s = &lds[tid * 27];
    s[ 0] = r_vrot.x;  s[ 1] = r_vrot.y;  s[ 2] = r_vrot.z;   // r_vis_rot
    s[ 3] = r_vt.x;    s[ 4] = r_vt.y;    s[ 5] = r_vt.z;     // r_vis_t
    s[ 6] = r_rot.x;   s[ 7] = r_rot.y;   s[ 8] = r_rot.z;    // r_rot
    s[ 9] = r_cross.x; s[10] = r_cross.y; s[11] = r_cross.z;  // r_cross
    s[12] = r_dv.x;    s[13] = r_dv.y;    s[14] = r_dv.z;     // r_deltav
    s[15] = d_ba.x;    s[16] = d_ba.y;    s[17] = d_ba.z;     // ba_kp1-ba_k
    s[18] = d_bg.x;    s[19] = d_bg.y;    s[20] = d_bg.z;     // bg_kp1-bg_k
    s[21] = ba_k.x;    s[22] = ba_k.y;    s[23] = ba_k.z;     // ba_k
    s[24] = bg_k.x;    s[25] = bg_k.y;    s[26] = bg_k.z;     // bg_k
  }
  __syncthreads();

  // Block-cooperative coalesced writeback: each wave store instruction covers
  // 128 contiguous bytes of the output.
  int count = N - base;
  if (count > THREADS) count = THREADS;
  if (count > 0) {
    const int total = count * 27;
    float* __restrict__ dst = out + 27ll * base;
    for (int k = tid; k < total; k += THREADS) {
      dst[k] = lds[k];
    }
  }
}

extern "C" void kernel_launch(void* const* d_in, const int* in_sizes, int n_in,
                              void* d_out, int out_size, void* d_ws, size_t ws_size,
                              hipStream_t stream) {
  const float* R0           = (const float*)d_in[0];
  const float* t0           = (const float*)d_in[1];
  const float* rotations    = (const float*)d_in[2];
  const float* translations = (const float*)d_in[3];
  const float* velocities   = (const float*)d_in[4];
  const float* bias_a       = (const float*)d_in[5];
  const float* bias_g       = (const float*)d_in[6];
  const float* dR           = (const float*)d_in[7];
  const float* dv           = (const float*)d_in[8];
  const float* dp           = (const float*)d_in[9];
  const float* dt           = (const float*)d_in[10];
  const float* vR           = (const float*)d_in[11];
  const float* vt           = (const float*)d_in[12];
  float* out = (float*)d_out;

  const int N = out_size / 27;  // W - 1 windows
  const int grid = (N + THREADS - 1) / THREADS;

  sliding_window_cost_kernel<<<grid, THREADS, 0, stream>>>(
      R0, t0, rotations, translations, velocities, bias_a, bias_g,
      dR, dv, dp, dt, vR, vt, out, N);
}